// MapGraphNet_16217796510182
// MI455X (gfx1250) — compile-verified
//
#include <hip/hip_runtime.h>
#include <math.h>

typedef float v2f __attribute__((ext_vector_type(2)));
typedef float v8f __attribute__((ext_vector_type(8)));

#define N_NODES 50000
#define N_EDGES 800000
#define OUT_DIM 394
#define HSTRIDE 400          // padded feature stride (zeros in cols 394..399)
#define WPAD_ROWS 416        // zero-padded fc_w rows (only 0..399 ever addressed)
#define M_TILES 3125         // 50000 / 16 exactly
#define CTILES 5             // col tiles per wave; grid.y=5 -> 25 tiles = ceil(394/16)

// ---------------- degree / normalization ----------------
__global__ void k_init_deg(float* __restrict__ deg) {
    int i = blockIdx.x * 256 + threadIdx.x;
    if (i < N_NODES) deg[i] = 1.0f;   // self-loop contributes 1
}

__global__ void k_deg_scatter(const long long* __restrict__ ei, float* __restrict__ deg) {
    int e = blockIdx.x * 256 + threadIdx.x;
    if (e < N_EDGES) {
        int dst = (int)ei[N_EDGES + e];
        atomicAdd(&deg[dst], 1.0f);
    }
}

__global__ void k_dinv(float* __restrict__ deg) {
    int i = blockIdx.x * 256 + threadIdx.x;
    if (i < N_NODES) deg[i] = rsqrtf(deg[i]);   // deg >= 1 always
}

// ---------------- acc init with self-loop message ----------------
// acc[n][f] = xl[n][f] * dinv[n]   (f < 394), 0 in pad columns
__global__ void k_init_acc(const float* __restrict__ x, const float* __restrict__ Wg,
                           const float* __restrict__ dinv, float* __restrict__ acc) {
    int n = blockIdx.x;
    float x0 = x[2 * n], x1 = x[2 * n + 1];
    float s = dinv[n];
    float* row = acc + (size_t)n * HSTRIDE;
    for (int f = threadIdx.x; f < HSTRIDE; f += 128) {
        float v = 0.0f;
        if (f < OUT_DIM) v = (x0 * Wg[2 * f] + x1 * Wg[2 * f + 1]) * s;
        row[f] = v;
    }
}

// ---------------- edge scatter: acc[dst] += xl[src]*dinv[src] ----------------
// Recompute the 394-wide message from the 8-byte x[src] instead of gathering a
// 1.6 KB xl row; the 80 MB accumulator is L2-resident (192 MB L2).
__global__ void k_edge_scatter(const long long* __restrict__ ei, const float* __restrict__ x,
                               const float* __restrict__ Wg, const float* __restrict__ dinv,
                               float* __restrict__ acc) {
    int e = blockIdx.x;
    int src = (int)ei[e];
    int dst = (int)ei[N_EDGES + e];
    float x0 = x[2 * src], x1 = x[2 * src + 1];
    float s = dinv[src];
    float* drow = acc + (size_t)dst * HSTRIDE;
    for (int f = threadIdx.x; f < OUT_DIM; f += 128) {
        float v = (x0 * Wg[2 * f] + x1 * Wg[2 * f + 1]) * s;
        atomicAdd(&drow[f], v);
    }
}

// ---------------- dst-normalize + bias + ReLU + BatchNorm (in place) ----------------
__global__ void k_bn_relu(float* __restrict__ acc, const float* __restrict__ dinv,
                          const float* __restrict__ b_gcn, const float* __restrict__ gamma,
                          const float* __restrict__ beta, const float* __restrict__ mean,
                          const float* __restrict__ var) {
    int n = blockIdx.x;
    float s = dinv[n];
    float* row = acc + (size_t)n * HSTRIDE;
    for (int f = threadIdx.x; f < OUT_DIM; f += 128) {
        float a = row[f] * s + b_gcn[f];
        a = fmaxf(a, 0.0f);
        float sc = gamma[f] * rsqrtf(var[f] + 1e-5f);
        row[f] = a * sc + (beta[f] - mean[f] * sc);
    }
}

// ---------------- zero-padded copy of fc_w ----------------
__global__ void k_pad_w(const float* __restrict__ fcw, float* __restrict__ wpad) {
    int idx = blockIdx.x * 256 + threadIdx.x;
    if (idx < WPAD_ROWS * HSTRIDE) {
        int j = idx / HSTRIDE, k = idx - j * HSTRIDE;
        wpad[idx] = (j < OUT_DIM && k < OUT_DIM) ? fcw[j * OUT_DIM + k] : 0.0f;
    }
}

// ---------------- FC GEMM: z = h @ fc_w.T + fc_b via V_WMMA_F32_16X16X4_F32 ----------------
// One wave: one 16-row M-tile x five 16-col N-tiles. Branch-free inner loop
// (zero padding makes out-of-range K harmless) with a register double-buffer so
// next K-group loads overlap the current group's 5 WMMAs.
// A frag (16x4): lane m = lane%16, K = k0 + 2*(lane/16) + {0,1}
// B frag (4x16): lane col = lane%16, same K pattern; B[k][j] = fc_w[j][k]
// C/D frag: VGPR v -> row v + 8*(lane/16), col lane%16
__global__ void __launch_bounds__(256) k_gemm(const float* __restrict__ h,
                                              const float* __restrict__ wpad,
                                              const float* __restrict__ fc_b,
                                              float* __restrict__ z) {
    int wave = threadIdx.x >> 5;
    int lane = threadIdx.x & 31;
    int mtile = blockIdx.x * 8 + wave;
    if (mtile >= M_TILES) return;                 // wave-uniform
    int laneHi = lane >> 4;
    int lane16 = lane & 15;
    int tbase = blockIdx.y * CTILES;              // first col tile (0,5,10,15,20)

    const float* hrow = h + (size_t)(mtile * 16 + lane16) * HSTRIDE + 2 * laneHi;
    const float* brow[CTILES];
    #pragma unroll
    for (int j = 0; j < CTILES; ++j)
        brow[j] = wpad + (size_t)((tbase + j) * 16 + lane16) * HSTRIDE + 2 * laneHi;

    v8f zero = {0.f, 0.f, 0.f, 0.f, 0.f, 0.f, 0.f, 0.f};
    v8f c[CTILES];
    #pragma unroll
    for (int j = 0; j < CTILES; ++j) c[j] = zero;

    // prologue: K-group 0 fragments
    float2 a = *(const float2*)(hrow);
    float2 b[CTILES];
    #pragma unroll
    for (int j = 0; j < CTILES; ++j) b[j] = *(const float2*)(brow[j]);

    for (int k0 = 0; k0 < HSTRIDE; k0 += 4) {
        // prefetch next K-group (final one is dead; buffers carry a tail margin)
        float2 an = *(const float2*)(hrow + k0 + 4);
        float2 bn[CTILES];
        #pragma unroll
        for (int j = 0; j < CTILES; ++j) bn[j] = *(const float2*)(brow[j] + k0 + 4);

        v2f av; av.x = a.x; av.y = a.y;
        #pragma unroll
        for (int j = 0; j < CTILES; ++j) {
            v2f bv; bv.x = b[j].x; bv.y = b[j].y;
            c[j] = __builtin_amdgcn_wmma_f32_16x16x4_f32(false, av, false, bv,
                                                         (short)0, c[j], false, false);
        }
        a = an;
        #pragma unroll
        for (int j = 0; j < CTILES; ++j) b[j] = bn[j];
    }

    int rbase = mtile * 16 + 8 * laneHi;
    #pragma unroll
    for (int j = 0; j < CTILES; ++j) {
        int col = (tbase + j) * 16 + lane16;
        if (col < OUT_DIM) {                       // only tile 24 is partial
            float bias = fc_b[col];
            #pragma unroll
            for (int v = 0; v < 8; ++v)
                z[(size_t)(rbase + v) * OUT_DIM + col] = c[j][v] + bias;
        }
    }
}

// ---------------- row-wise log_softmax (one wave32 per row) ----------------
__global__ void k_logsoftmax(const float* __restrict__ z, float* __restrict__ out) {
    int wave = threadIdx.x >> 5;
    int lane = threadIdx.x & 31;
    int row = blockIdx.x * 8 + wave;              // 6250 * 8 = 50000 exact
    const float* zr = z + (size_t)row * OUT_DIM;
    float m = -3.4e38f;
    for (int c = lane; c < OUT_DIM; c += 32) m = fmaxf(m, zr[c]);
    #pragma unroll
    for (int o = 16; o >= 1; o >>= 1) m = fmaxf(m, __shfl_xor(m, o, 32));
    float s = 0.0f;
    for (int c = lane; c < OUT_DIM; c += 32) s += expf(zr[c] - m);
    #pragma unroll
    for (int o = 16; o >= 1; o >>= 1) s += __shfl_xor(s, o, 32);
    float lg = m + logf(s);
    float* orow = out + (size_t)row * OUT_DIM;
    for (int c = lane; c < OUT_DIM; c += 32) orow[c] = zr[c] - lg;
}

extern "C" void kernel_launch(void* const* d_in, const int* in_sizes, int n_in,
                              void* d_out, int out_size, void* d_ws, size_t ws_size,
                              hipStream_t stream) {
    const float*     x     = (const float*)d_in[0];
    const long long* ei    = (const long long*)d_in[1];
    const float*     Wg    = (const float*)d_in[2];
    const float*     b_gcn = (const float*)d_in[3];
    const float*     gamma = (const float*)d_in[4];
    const float*     beta  = (const float*)d_in[5];
    const float*     mean  = (const float*)d_in[6];
    const float*     var   = (const float*)d_in[7];
    const float*     fcw   = (const float*)d_in[8];
    const float*     fcb   = (const float*)d_in[9];

    float* lsm = (float*)d_out;                               // output 0: log_softmax(z)
    float* z   = (float*)d_out + (size_t)N_NODES * OUT_DIM;   // output 1: z

    float* ws   = (float*)d_ws;
    float* dinv = ws;                                         // 50000 floats
    float* acc  = ws + 50048;                                 // 50000*400 floats (80 MB)
    // +64-float margin after acc so the final dead prefetch can't fault
    float* wpad = ws + 50048 + (size_t)N_NODES * HSTRIDE + 64;  // 416*400 floats (+margin)

    k_init_deg    <<<(N_NODES + 255) / 256, 256, 0, stream>>>(dinv);
    k_deg_scatter <<<(N_EDGES + 255) / 256, 256, 0, stream>>>(ei, dinv);
    k_dinv        <<<(N_NODES + 255) / 256, 256, 0, stream>>>(dinv);
    k_pad_w       <<<(WPAD_ROWS * HSTRIDE + 255) / 256, 256, 0, stream>>>(fcw, wpad);
    k_init_acc    <<<N_NODES, 128, 0, stream>>>(x, Wg, dinv, acc);
    k_edge_scatter<<<N_EDGES, 128, 0, stream>>>(ei, x, Wg, dinv, acc);
    k_bn_relu     <<<N_NODES, 128, 0, stream>>>(acc, dinv, b_gcn, gamma, beta, mean, var);
    k_gemm        <<<dim3((M_TILES + 7) / 8, CTILES), 256, 0, stream>>>(acc, wpad, fcb, z);
    k_logsoftmax  <<<N_NODES / 8, 256, 0, stream>>>(z, lsm);
}